// GEM_2010044694628
// MI455X (gfx1250) — compile-verified
//
#include <hip/hip_runtime.h>
#include <hip/hip_bf16.h>
#include <math.h>

typedef __attribute__((ext_vector_type(2))) float v2f;
typedef __attribute__((ext_vector_type(8))) float v8f;

static __device__ __forceinline__ v8f wmma_f32(v2f a, v2f b, v8f c) {
    return __builtin_amdgcn_wmma_f32_16x16x4_f32(
               false, a, false, b, (short)0, c, false, false);
}

// Load one K-group (A fragment + 4 B fragments) for the GEMM pipeline.
static __device__ __forceinline__ void ld_gemm_group(
    const float* __restrict__ Ap, const float* const* Wp, int kk,
    v2f& a, v2f* b)
{
    a.x = Ap[kk]; a.y = Ap[kk + 1];
#pragma unroll
    for (int j = 0; j < 4; ++j) { b[j].x = Wp[j][kk]; b[j].y = Wp[j][kk + 1]; }
}

// ---------------------------------------------------------------------------
// GEMM + bias:  C[M,Nc] = A[M,K] @ W[Nc,K]^T + bias[Nc]
// Each wave computes a 16x64 C tile (4 accumulators share one A fragment).
// The K loop is software-pipelined depth-2: while the wmmas of group k run,
// the 5 loads of group k+8 are in flight (loads retire in order, so each
// group's wait leaves the next group's loads outstanding).
// blockDim = 128 (4 waves, 256 cols/block); grid.x = Nc/256, grid.y = ceil(M/16).
// ---------------------------------------------------------------------------
__global__ __launch_bounds__(128) void gemm_bias_wmma(
    const float* __restrict__ A, const float* __restrict__ W,
    const float* __restrict__ bias, float* __restrict__ C,
    int Mrows, int Ncols, int K)
{
    const int lane = threadIdx.x & 31;
    const int wave = threadIdx.x >> 5;
    const int l    = lane & 15;
    const int half = lane >> 4;

    const int colBase = (blockIdx.x * 4 + wave) * 64;
    const int rowBase = blockIdx.y * 16;
    if (colBase >= Ncols) return;                 // wave-uniform

    int arow = rowBase + l; if (arow >= Mrows) arow = Mrows - 1;
    const float* __restrict__ Ap = A + (size_t)arow * K;

    const float* Wp[4];
#pragma unroll
    for (int j = 0; j < 4; ++j)
        Wp[j] = W + (size_t)(colBase + j * 16 + l) * K;

    // gfx1250 speculative prefetch of the A stream (global_prefetch_b8)
    __builtin_prefetch(Ap, 0, 1);
    __builtin_prefetch(Ap + 64, 0, 1);

    v8f acc[4] = {};
    v2f a0, a1, b0[4], b1[4];
    const int kb = 2 * half;                      // A 16x4 layout: half picks K pair
    ld_gemm_group(Ap, Wp, kb,     a0, b0);        // group k=0
    ld_gemm_group(Ap, Wp, kb + 4, a1, b1);        // group k=4

    int k0 = 0;
    for (; k0 <= K - 16; k0 += 8) {
#pragma unroll
        for (int j = 0; j < 4; ++j) acc[j] = wmma_f32(a0, b0[j], acc[j]);
        ld_gemm_group(Ap, Wp, k0 + 8 + kb, a0, b0);
#pragma unroll
        for (int j = 0; j < 4; ++j) acc[j] = wmma_f32(a1, b1[j], acc[j]);
        ld_gemm_group(Ap, Wp, k0 + 12 + kb, a1, b1);
    }
#pragma unroll
    for (int j = 0; j < 4; ++j) acc[j] = wmma_f32(a0, b0[j], acc[j]);
#pragma unroll
    for (int j = 0; j < 4; ++j) acc[j] = wmma_f32(a1, b1[j], acc[j]);

#pragma unroll
    for (int j = 0; j < 4; ++j) {
        const int col = colBase + j * 16 + l;
        const float bcol = bias[col];
#pragma unroll
        for (int r = 0; r < 8; ++r) {
            const int row = rowBase + r + 8 * half;   // C/D: VGPR r -> row r+8*half
            if (row < Mrows)
                C[(size_t)row * Ncols + col] = acc[j][r] + bcol;
        }
    }
}

// K-group loader for attention phase 1 (A from Q, 2 B fragments from K-mat).
static __device__ __forceinline__ void ld_attn1_group(
    const float* __restrict__ qp, const float* __restrict__ kp0,
    const float* __restrict__ kp1, int kk, v2f& a, v2f& b0, v2f& b1)
{
    a.x  = qp[kk];  a.y  = qp[kk + 1];
    b0.x = kp0[kk]; b0.y = kp0[kk + 1];
    b1.x = kp1[kk]; b1.y = kp1[kk + 1];
}

// K-group loader for attention phase 3 (A from LDS tile P, 4 B from V).
static __device__ __forceinline__ void ld_attn3_group(
    const float* __restrict__ Prow, const float* __restrict__ v,
    const int* col, int kk, v2f& a, v2f* b)
{
    a.x = Prow[kk]; a.y = Prow[kk + 1];
    int vk0 = kk;     if (vk0 > 48) vk0 = 48;     // padded P cols are 0; clamp
    int vk1 = kk + 1; if (vk1 > 48) vk1 = 48;     // keeps V loads in-bounds
    const float* __restrict__ v0 = v + (size_t)vk0 * 512;
    const float* __restrict__ v1 = v + (size_t)vk1 * 512;
#pragma unroll
    for (int j = 0; j < 4; ++j) { b[j].x = v0[col[j]]; b[j].y = v1[col[j]]; }
}

// ---------------------------------------------------------------------------
// Attention block (49 query tokens x 49 key tokens, H=256, Cv=512):
//   P = softmax(Q @ Kmat^T * scale);  Out = P @ V
// mode 0 (FAM): blockIdx.x = anchor n; Kmat/V shared (sel 0)
// mode 1 (ARM): blockIdx.x = pair p = i*32+j; query sel j = p%32, key/val i = p/32
// 256 threads = 8 waves; 64x64 LDS score tile, cols 49..63 zeroed post-softmax
// so the K=64 second WMMA pass is exact.
// ---------------------------------------------------------------------------
__global__ __launch_bounds__(256) void attn_wmma(
    const float* __restrict__ Qb, const float* __restrict__ Kb,
    const float* __restrict__ Vb, float* __restrict__ Outb,
    int mode, float scale)
{
    __shared__ float P[64 * 64];                  // 16 KB

    int qsel, ksel;
    if (mode == 0) { qsel = blockIdx.x; ksel = 0; }
    else           { qsel = blockIdx.x & 31; ksel = blockIdx.x >> 5; }

    const float* __restrict__ q    = Qb + (size_t)qsel * 49 * 256;
    const float* __restrict__ kmat = Kb + (size_t)ksel * 49 * 256;
    const float* __restrict__ v    = Vb + (size_t)ksel * 49 * 512;
    float* __restrict__ out        = Outb + (size_t)blockIdx.x * 49 * 512;

    const int lane = threadIdx.x & 31;
    const int wave = threadIdx.x >> 5;
    const int l    = lane & 15;
    const int half = lane >> 4;
    const int tm   = wave >> 1;                   // row tile 0..3 (2 waves each)
    const int kb   = 2 * half;

    // ---- Phase 1: scores = Q @ K^T * scale; wave owns 16x32 (2 col tiles) --
    {
        const int tn0 = (wave & 1) * 2;
        int drow  = tm * 16 + l;         if (drow  > 48) drow  = 48;
        int erow0 = tn0 * 16 + l;        if (erow0 > 48) erow0 = 48;
        int erow1 = (tn0 + 1) * 16 + l;  if (erow1 > 48) erow1 = 48;
        const float* __restrict__ qp  = q    + (size_t)drow  * 256;
        const float* __restrict__ kp0 = kmat + (size_t)erow0 * 256;
        const float* __restrict__ kp1 = kmat + (size_t)erow1 * 256;
        __builtin_prefetch(qp, 0, 1);

        v8f acc0 = {}, acc1 = {};
        v2f a0, a1, c0, c1, d0, d1;
        ld_attn1_group(qp, kp0, kp1, kb,     a0, c0, d0);
        ld_attn1_group(qp, kp0, kp1, kb + 4, a1, c1, d1);
        int k0 = 0;
        for (; k0 <= 256 - 16; k0 += 8) {
            acc0 = wmma_f32(a0, c0, acc0);
            acc1 = wmma_f32(a0, d0, acc1);
            ld_attn1_group(qp, kp0, kp1, k0 + 8 + kb, a0, c0, d0);
            acc0 = wmma_f32(a1, c1, acc0);
            acc1 = wmma_f32(a1, d1, acc1);
            ld_attn1_group(qp, kp0, kp1, k0 + 12 + kb, a1, c1, d1);
        }
        acc0 = wmma_f32(a0, c0, acc0);
        acc1 = wmma_f32(a0, d0, acc1);
        acc0 = wmma_f32(a1, c1, acc0);
        acc1 = wmma_f32(a1, d1, acc1);

#pragma unroll
        for (int r = 0; r < 8; ++r) {
            const int prow = (tm * 16 + r + 8 * half) * 64;
            P[prow + tn0 * 16 + l]       = acc0[r] * scale;
            P[prow + (tn0 + 1) * 16 + l] = acc1[r] * scale;
        }
    }
    __syncthreads();

    // ---- Phase 2: row-wise softmax over 49 valid cols; zero-pad 49..63 ----
    if (threadIdx.x < 49) {
        float* row = P + threadIdx.x * 64;
        float m = row[0];
        for (int e = 1; e < 49; ++e) m = fmaxf(m, row[e]);
        float s = 0.f;
        for (int e = 0; e < 49; ++e) { float ex = __expf(row[e] - m); row[e] = ex; s += ex; }
        const float inv = 1.f / s;
        for (int e = 0; e < 49; ++e) row[e] *= inv;
        for (int e = 49; e < 64; ++e) row[e] = 0.f;
    }
    __syncthreads();

    // ---- Phase 3: Out = P @ V; wave owns row tile tm, groups of 4 col tiles
    {
        const float* __restrict__ Prow = P + (tm * 16 + l) * 64;
        for (int g0 = (wave & 1) * 4; g0 < 32; g0 += 8) {
            int col[4];
#pragma unroll
            for (int j = 0; j < 4; ++j) col[j] = (g0 + j) * 16 + l;

            v8f acc[4] = {};
            v2f a0, a1, b0[4], b1[4];
            ld_attn3_group(Prow, v, col, kb,     a0, b0);
            ld_attn3_group(Prow, v, col, kb + 4, a1, b1);
            int k0 = 0;
            for (; k0 <= 64 - 16; k0 += 8) {
#pragma unroll
                for (int j = 0; j < 4; ++j) acc[j] = wmma_f32(a0, b0[j], acc[j]);
                ld_attn3_group(Prow, v, col, k0 + 8 + kb, a0, b0);
#pragma unroll
                for (int j = 0; j < 4; ++j) acc[j] = wmma_f32(a1, b1[j], acc[j]);
                ld_attn3_group(Prow, v, col, k0 + 12 + kb, a1, b1);
            }
#pragma unroll
            for (int j = 0; j < 4; ++j) acc[j] = wmma_f32(a0, b0[j], acc[j]);
#pragma unroll
            for (int j = 0; j < 4; ++j) acc[j] = wmma_f32(a1, b1[j], acc[j]);

#pragma unroll
            for (int j = 0; j < 4; ++j) {
#pragma unroll
                for (int r = 0; r < 8; ++r) {
                    const int row = tm * 16 + r + 8 * half;
                    if (row < 49)
                        out[(size_t)row * 512 + col[j]] = acc[j][r];
                }
            }
        }
    }
}

// ---------------------------------------------------------------------------
// Per-pair BatchNorm (training stats, gamma=1, beta=0): block = one pair,
// reduce 49*512 elements, then normalize in place. d_out (103 MB) fits in the
// 192 MB L2, so the second pass is L2-resident.
// ---------------------------------------------------------------------------
__global__ __launch_bounds__(256) void bn_pair(float* __restrict__ edge)
{
    __shared__ float ssum[256];
    __shared__ float ssq[256];
    const int n = 49 * 512;
    float* __restrict__ p = edge + (size_t)blockIdx.x * n;

    float s = 0.f, sq = 0.f;
    for (int i = threadIdx.x; i < n; i += 256) {
        const float x = p[i];
        s += x; sq += x * x;
    }
    ssum[threadIdx.x] = s; ssq[threadIdx.x] = sq;
    __syncthreads();
    for (int off = 128; off > 0; off >>= 1) {
        if (threadIdx.x < off) {
            ssum[threadIdx.x] += ssum[threadIdx.x + off];
            ssq[threadIdx.x]  += ssq[threadIdx.x + off];
        }
        __syncthreads();
    }
    const float mean = ssum[0] * (1.f / n);
    const float var  = ssq[0] * (1.f / n) - mean * mean;
    const float inv  = rsqrtf(var + 1e-5f);
    for (int i = threadIdx.x; i < n; i += 256)
        p[i] = (p[i] - mean) * inv;
}

// ---------------------------------------------------------------------------
extern "C" void kernel_launch(void* const* d_in, const int* in_sizes, int n_in,
                              void* d_out, int out_size, void* d_ws, size_t ws_size,
                              hipStream_t stream)
{
    (void)in_sizes; (void)n_in; (void)out_size; (void)ws_size;

    const float* cf     = (const float*)d_in[0];   // [32,49,512]
    const float* gf     = (const float*)d_in[1];   // [49,512]
    const float* gft_w  = (const float*)d_in[2];   // [512,512]
    const float* gft_b  = (const float*)d_in[3];
    const float* fam_qw = (const float*)d_in[4];   // [256,512]
    const float* fam_qb = (const float*)d_in[5];
    const float* fam_kw = (const float*)d_in[6];
    const float* fam_kb = (const float*)d_in[7];
    const float* fam_vw = (const float*)d_in[8];   // [512,512]
    const float* fam_vb = (const float*)d_in[9];
    const float* arm_qw = (const float*)d_in[10];
    const float* arm_qb = (const float*)d_in[11];
    const float* arm_kw = (const float*)d_in[12];
    const float* arm_kb = (const float*)d_in[13];
    const float* arm_vw = (const float*)d_in[14];
    const float* arm_vb = (const float*)d_in[15];
    const float* ep_w   = (const float*)d_in[16];
    const float* ep_b   = (const float*)d_in[17];

    float* out = (float*)d_out;                    // [1024,49,512]

    // workspace layout (floats); total ~114 MB
    float* w = (float*)d_ws;
    float* g    = w;               w += 49 * 512;
    float* q    = w;               w += 32 * 49 * 256;
    float* kk   = w;               w += 49 * 256;
    float* v    = w;               w += 49 * 512;
    float* feat = w;               w += 32 * 49 * 512;
    float* qa   = w;               w += 32 * 49 * 256;
    float* ka   = w;               w += 32 * 49 * 256;
    float* va   = w;               w += 32 * 49 * 512;
    float* pf   = w;               w += 1024 * 49 * 512;

    const float scale = 0.0625f;   // 1/sqrt(H=256)

    // 1) g = gf @ gft_w^T + gft_b           [49,512]
    gemm_bias_wmma<<<dim3(2, 4), 128, 0, stream>>>(gf, gft_w, gft_b, g, 49, 512, 512);
    // 2) q = cf @ fam_qw^T + fam_qb         [1568,256]
    gemm_bias_wmma<<<dim3(1, 98), 128, 0, stream>>>(cf, fam_qw, fam_qb, q, 1568, 256, 512);
    // 3) k = g @ fam_kw^T + fam_kb          [49,256]
    gemm_bias_wmma<<<dim3(1, 4), 128, 0, stream>>>(g, fam_kw, fam_kb, kk, 49, 256, 512);
    // 4) v = g @ fam_vw^T + fam_vb          [49,512]
    gemm_bias_wmma<<<dim3(2, 4), 128, 0, stream>>>(g, fam_vw, fam_vb, v, 49, 512, 512);
    // 5) FAM attention -> feat              [32,49,512]
    attn_wmma<<<32, 256, 0, stream>>>(q, kk, v, feat, 0, scale);
    // 6) ARM projections from feat
    gemm_bias_wmma<<<dim3(1, 98), 128, 0, stream>>>(feat, arm_qw, arm_qb, qa, 1568, 256, 512);
    gemm_bias_wmma<<<dim3(1, 98), 128, 0, stream>>>(feat, arm_kw, arm_kb, ka, 1568, 256, 512);
    gemm_bias_wmma<<<dim3(2, 98), 128, 0, stream>>>(feat, arm_vw, arm_vb, va, 1568, 512, 512);
    // 7) ARM pair attention -> pair_feat    [1024,49,512]
    attn_wmma<<<1024, 256, 0, stream>>>(qa, ka, va, pf, 1, scale);
    // 8) edge = pf @ ep_w^T + ep_b -> d_out [50176,512]  (26.3 GFLOP, dominant)
    gemm_bias_wmma<<<dim3(2, 3136), 128, 0, stream>>>(pf, ep_w, ep_b, out, 50176, 512, 512);
    // 9) per-pair BatchNorm in place
    bn_pair<<<1024, 256, 0, stream>>>(out);
}